// GraphConvolution_26912265076934
// MI455X (gfx1250) — compile-verified
//
#include <hip/hip_runtime.h>

#define N_NODES 100000
#define N_EDGES 1280000
#define IN_DIM  256
#define OUT_DIM 64

typedef float v2f __attribute__((ext_vector_type(2)));
typedef float v8f __attribute__((ext_vector_type(8)));

// ---------------------------------------------------------------------------
// Kernel 1: pre_sup = X @ W using V_WMMA_F32_16X16X4_F32.
// Block = 256 threads = 8 waves; wave (mtile,ntile) computes a 16x16 tile.
// Block tile: 32 rows x 64 cols. Grid = 100000/32 = 3125 (exact, no tail).
// W (64 KB) staged in LDS once per block.
// ---------------------------------------------------------------------------
__global__ __launch_bounds__(256) void gcn_gemm_wmma(
    const float* __restrict__ X, const float* __restrict__ W,
    float* __restrict__ P)
{
    __shared__ float sW[IN_DIM * OUT_DIM];          // 64 KB LDS
    {
        const float4* Wv  = (const float4*)W;
        float4*       sWv = (float4*)sW;
        #pragma unroll
        for (int i = threadIdx.x; i < (IN_DIM * OUT_DIM) / 4; i += 256)
            sWv[i] = Wv[i];
    }
    __syncthreads();

    const int wave    = threadIdx.x >> 5;
    const int lane    = threadIdx.x & 31;
    const int mtile   = wave >> 2;                  // 0..1
    const int ntile   = wave & 3;                   // 0..3
    const int halfsel = lane >> 4;                  // 0 or 1 (lane half)
    const int l16     = lane & 15;

    const int rowBase = blockIdx.x * 32 + mtile * 16;
    const int m       = rowBase + l16;              // A-matrix row for this lane
    const int ncol    = ntile * 16 + l16;           // B/C column for this lane
    const int koff    = halfsel * 2;                // K offset per ISA A/B layout

    // A 16x4 layout: lane(l16, half) holds {A[m][k0+koff], A[m][k0+koff+1]}
    const float* xrow = X + (size_t)m * IN_DIM + koff;
    // B 4x16 layout: lane holds {B[k0+koff][ncol], B[k0+koff+1][ncol]}
    const float* wcol = sW + (size_t)koff * OUT_DIM + ncol;

    v8f acc = {0.f, 0.f, 0.f, 0.f, 0.f, 0.f, 0.f, 0.f};

    #pragma unroll 8
    for (int k0 = 0; k0 < IN_DIM; k0 += 4) {
        v2f a;
        a.x = xrow[k0];
        a.y = xrow[k0 + 1];
        v2f b;
        b.x = wcol[(k0)     * OUT_DIM];
        b.y = wcol[(k0 + 1) * OUT_DIM];
        // (neg_a, A, neg_b, B, c_mod, C, reuse_a, reuse_b)
        acc = __builtin_amdgcn_wmma_f32_16x16x4_f32(
            false, a, false, b, (short)0, acc, false, false);
    }

    // D 16x16 f32 layout: VGPR r -> row rowBase + r + 8*halfsel, col ncol
    float* outp = P + ((size_t)(rowBase + halfsel * 8)) * OUT_DIM + ncol;
    #pragma unroll
    for (int r = 0; r < 8; ++r)
        outp[(size_t)r * OUT_DIM] = acc[r];
}

// ---------------------------------------------------------------------------
// Kernel 2: zero the accumulation buffer (d_out).
// ---------------------------------------------------------------------------
__global__ void gcn_zero(float4* __restrict__ out, int n4)
{
    int i = blockIdx.x * blockDim.x + threadIdx.x;
    if (i < n4) out[i] = make_float4(0.f, 0.f, 0.f, 0.f);
}

// ---------------------------------------------------------------------------
// Kernel 3: one edge per wave32. Each lane handles 2 of the 64 output floats:
// gather pre_sup[col], scale by val, atomic-add into out[row].
// pre_sup and out both fit in the 192 MB L2, so atomics resolve in L2.
// ---------------------------------------------------------------------------
__global__ __launch_bounds__(256) void gcn_edge_scatter(
    const int* __restrict__ rows, const int* __restrict__ cols,
    const float* __restrict__ vals, const float* __restrict__ P,
    float* __restrict__ out)
{
    const int e = blockIdx.x * 8 + (threadIdx.x >> 5);
    if (e >= N_EDGES) return;
    const int   lane = threadIdx.x & 31;
    const int   r    = rows[e];
    const int   c    = cols[e];
    const float v    = vals[e];

    const float2 p = *((const float2*)(P + (size_t)c * OUT_DIM) + lane);
    float* dst = out + (size_t)r * OUT_DIM + lane * 2;
    atomicAdd(dst,     p.x * v);
    atomicAdd(dst + 1, p.y * v);
}

// ---------------------------------------------------------------------------
// Kernel 4: out = relu(out + bias)
// ---------------------------------------------------------------------------
__global__ void gcn_finalize(float* __restrict__ out,
                             const float* __restrict__ bias)
{
    size_t i = (size_t)blockIdx.x * blockDim.x + threadIdx.x;
    if (i < (size_t)N_NODES * OUT_DIM) {
        float x = out[i] + bias[i & (OUT_DIM - 1)];
        out[i] = x > 0.f ? x : 0.f;
    }
}

extern "C" void kernel_launch(void* const* d_in, const int* in_sizes, int n_in,
                              void* d_out, int out_size, void* d_ws, size_t ws_size,
                              hipStream_t stream)
{
    const float* x    = (const float*)d_in[0];
    const int*   rows = (const int*)  d_in[1];
    const int*   cols = (const int*)  d_in[2];
    const float* vals = (const float*)d_in[3];
    const float* w    = (const float*)d_in[4];
    const float* bias = (const float*)d_in[5];
    float*       out  = (float*)d_out;
    float*       P    = (float*)d_ws;   // pre_sup: 100000*64*4 = 25.6 MB

    // 1) pre_sup = X @ W  (WMMA f32)
    gcn_gemm_wmma<<<N_NODES / 32, 256, 0, stream>>>(x, w, P);

    // 2) zero accumulator
    const int n4 = (N_NODES * OUT_DIM) / 4;
    gcn_zero<<<(n4 + 255) / 256, 256, 0, stream>>>((float4*)d_out, n4);

    // 3) gather * val -> atomic scatter-add
    gcn_edge_scatter<<<(N_EDGES + 7) / 8, 256, 0, stream>>>(rows, cols, vals, P, out);

    // 4) bias + relu
    const int ntot = N_NODES * OUT_DIM;
    gcn_finalize<<<(ntot + 255) / 256, 256, 0, stream>>>(out, bias);
}